// ReconDetectionConfidenceMap_42674795053847
// MI455X (gfx1250) — compile-verified
//
#include <hip/hip_runtime.h>
#include <hip/hip_bf16.h>
#include <math.h>

// Problem constants (match reference)
#define BB 32
#define KK 64
#define HH 128
#define WW 128
#define EPSK 1e-5f

// LDS score tile: per wave, [K][32 pixels]; stride 36 floats:
//  - row stride 144 B keeps 16B alignment for ds_store_b128
//  - 36*l15 mod 64 hits 16 distinct banks (36/4 = 9 is odd)
#define GSTRIDE 36
#define GS_PER_WAVE (KK * GSTRIDE)

typedef float v2f __attribute__((ext_vector_type(2)));
typedef float v4f __attribute__((ext_vector_type(4)));
typedef float v8f __attribute__((ext_vector_type(8)));

__global__ __launch_bounds__(128) void recon_detmap_kernel(
    const float* __restrict__ keypoints,  // [B,K,2] (x,y)
    const float* __restrict__ sxp,        // scalar
    const float* __restrict__ syp,        // scalar
    const float* __restrict__ rhop,       // scalar
    float* __restrict__ out)              // [B,K,H,W]
{
    // B-matrix operands pre-arranged per lane-half: index = half*KK + k
    __shared__ float B0L[2 * KK];         // half 0: ky[k]   half 1: 1.0
    __shared__ float B1L[2 * KK];         // half 0: kx[k]   half 1: v[k]
    __shared__ __align__(16) float gbuf[4 * GS_PER_WAVE];   // 36 KB

    const int y    = blockIdx.x;   // row
    const int b    = blockIdx.y;   // batch
    const int tid  = threadIdx.x;
    const int lane = tid & 31;
    const int wave = tid >> 5;
    const int l15  = lane & 15;
    const int half = lane >> 4;    // 0 = lanes 0-15, 1 = lanes 16-31

    // Uniform scalar parameters (fold log2(e) so exp -> raw v_exp_f32)
    const float sx   = sxp[0];
    const float sy   = syp[0];
    const float rho  = rhop[0];
    const float omr2 = 1.0f - rho * rho;
    const float num1 = -1.0f / (2.0f * omr2);
    const float L2E  = 1.44269504088896340736f;
    const float P = num1 * L2E / (sy * sy);
    const float Q = num1 * L2E / (sx * sx);
    const float R = -2.0f * rho * num1 * L2E / (sx * sy);
    const float den = 1.0f / (2.0f * 3.14159265358979323846f * sx * sy * sqrtf(omr2));

    // Per-block keypoint features into LDS, pre-swizzled for B layout:
    //   B 4x16 f32 operand: VGPR0 = rows {0,2}, VGPR1 = rows {1,3} (low/high half)
    //   row0 = ky, row1 = kx, row2 = 1, row3 = v_k
    if (tid < KK) {
        const float kx = keypoints[((size_t)b * KK + tid) * 2 + 0];
        const float ky = keypoints[((size_t)b * KK + tid) * 2 + 1];
        B0L[tid]      = ky;
        B0L[KK + tid] = 1.0f;
        B1L[tid]      = kx;
        B1L[KK + tid] = P * ky * ky + Q * kx * kx + R * kx * ky;
    }
    __syncthreads();

    // ---------------- Phase 1: bilinear exponent via V_WMMA_F32_16X16X4_F32 ----
    // g(pix,k) = a(pix) . b(k)
    //   a = [-2P*y - R*x, -2Q*x - R*y, P*y^2 + Q*x^2 + R*x*y, 1]
    //   b = [ky, kx, 1, P*ky^2 + Q*kx^2 + R*kx*ky]
    const float fy = (float)y;
    float* gw = &gbuf[wave * GS_PER_WAVE];

    // Load the four 16-keypoint B tiles once (unconditional ds loads, no selects)
    v2f Bt[4];
    #pragma unroll
    for (int kt = 0; kt < 4; ++kt) {
        const int bi = half * KK + kt * 16 + l15;
        Bt[kt].x = B0L[bi];
        Bt[kt].y = B1L[bi];
    }

    #pragma unroll
    for (int t = 0; t < 2; ++t) {                 // two 16-pixel subtiles per wave
        const float fx = (float)(wave * 32 + t * 16 + l15);
        const float a0 = -2.0f * P * fy - R * fx;
        const float a1 = -2.0f * Q * fx - R * fy;
        const float a2 = P * fy * fy + Q * fx * fx + R * fx * fy;
        // A 16x4 f32 operand: VGPR0 = K{0,2}, VGPR1 = K{1,3} (low/high lane half)
        v2f A;
        A.x = half ? a2 : a0;
        A.y = half ? 1.0f : a1;

        #pragma unroll
        for (int kt = 0; kt < 4; ++kt) {          // four 16-keypoint tiles (K=64)
            v8f D = {};
            D = __builtin_amdgcn_wmma_f32_16x16x4_f32(
                    false, A, false, Bt[kt], (short)0, D, false, false);

            // C/D layout: lane half selects M rows {v}/{v+8}, N = l15.
            // D[0..7] -> 8 consecutive pixels => two b128 LDS stores.
            float* dst = &gw[(kt * 16 + l15) * GSTRIDE + t * 16 + half * 8];
            v4f lo = { D[0], D[1], D[2], D[3] };
            v4f hi = { D[4], D[5], D[6], D[7] };
            *(v4f*)(dst)     = lo;
            *(v4f*)(dst + 4) = hi;
        }
    }
    __syncthreads();

    // ---------------- Phase 2: exp, K-normalize, coalesced NT stores ----------
    const int x = wave * 32 + lane;               // this lane's pixel column
    float s[KK];                                  // scores stay in VGPRs
    float denom = EPSK;
    #pragma unroll
    for (int k = 0; k < KK; ++k) {
        s[k] = den * __builtin_amdgcn_exp2f(gw[k * GSTRIDE + lane]);
        denom += s[k];
    }
    const float inv = 1.0f / denom;

    float* outp = out + (((size_t)b * KK) * HH + (size_t)y) * WW + x;
    #pragma unroll
    for (int k = 0; k < KK; ++k) {
        // streamed 128 MB output, no reuse -> non-temporal store
        __builtin_nontemporal_store(s[k] * inv, outp + (size_t)k * (HH * WW));
    }
}

extern "C" void kernel_launch(void* const* d_in, const int* in_sizes, int n_in,
                              void* d_out, int out_size, void* d_ws, size_t ws_size,
                              hipStream_t stream) {
    // setup_inputs order:
    // 0: batch_size (int scalar, unused)
    // 1: keypoints  [B,K,2] f32
    // 2: reconDetectionMap_std_x (f32 scalar)
    // 3: reconDetectionMap_std_y (f32 scalar)
    // 4: correlation (f32 scalar)
    // 5: DetectionMap [B,K,H,W] f32 (only its SHAPE is used by the reference)
    (void)in_sizes; (void)n_in; (void)d_ws; (void)ws_size; (void)out_size;

    const float* keypoints = (const float*)d_in[1];
    const float* sxp       = (const float*)d_in[2];
    const float* syp       = (const float*)d_in[3];
    const float* rhop      = (const float*)d_in[4];
    float* out             = (float*)d_out;

    dim3 grid(HH, BB);   // one block per (y, b)
    dim3 block(128);     // 4 waves of 32; each wave owns 32 x-pixels
    recon_detmap_kernel<<<grid, block, 0, stream>>>(keypoints, sxp, syp, rhop, out);
}